// AVWGCN_10694468567659
// MI455X (gfx1250) — compile-verified
//
#include <hip/hip_runtime.h>
#include <hip/hip_bf16.h>
#include <cstddef>

// ---------------------------------------------------------------------------
// AVWGCN pipeline for MI455X (gfx1250, wave32, WMMA).
// B=8, N=1024, DIM_IN=DIM_OUT=64, EMBED=16, DP=16.
//
//  K1a simraw  : R = relu(E E^T)                       [N,N]
//  K1b softmax : sim = row-softmax(R) in place
//  K2a cdist   : sub_e[b,i,j] = exp(-L1(pa[b,i],pa[b,j]))  (exact softmax
//                numerator: column max of -d1 is 0 via the diagonal)
//  K2b colsum  : S[b,j] = sum_i sub_e[b,i,j]
//  K2c xnorm   : xn = x / S   (folds softmax denominator into x)
//  K5  weights : W[n] = sum_d E[n,d] * Wp[d], stored K-pair-interleaved
//  K3  bmm     : sx  = sub_e @ xn    (f32 WMMA, per batch)
//  K4  bmm     : xg1 = sim  @ sx     (same kernel, A-stride 0)
//  K6  final   : out[b,n,:] = [sx|xg1][b,n,:] @ [W_n0;W_n1] + E[n]@bias_pool
// ---------------------------------------------------------------------------

typedef float v2f __attribute__((ext_vector_type(2)));
typedef float v8f __attribute__((ext_vector_type(8)));

static __device__ __forceinline__ v8f wmma_f32(v2f a, v2f b, v8f c) {
  // D = A(16x4, f32) * B(4x16, f32) + C(16x16, f32)
  return __builtin_amdgcn_wmma_f32_16x16x4_f32(
      /*neg_a=*/false, a, /*neg_b=*/false, b,
      /*c_mod=*/(short)0, c, /*reuse_a=*/false, /*reuse_b=*/false);
}

// ---------------- K1a: R = relu(E E^T), [1024,1024] -------------------------
__global__ void k_simraw(const float* __restrict__ E, float* __restrict__ sim) {
  const int idx = blockIdx.x * 256 + threadIdx.x;   // < 1024*1024
  const int i = idx >> 10, j = idx & 1023;
  float acc = 0.f;
#pragma unroll
  for (int d = 0; d < 16; ++d) acc += E[i * 16 + d] * E[j * 16 + d];
  sim[idx] = fmaxf(acc, 0.f);
}

// ---------------- K1b: row softmax in place ---------------------------------
__global__ void k_rowsoftmax(float* __restrict__ sim) {
  const int row = blockIdx.x;
  const int t = threadIdx.x;                         // 256 threads, 4 elems each
  __shared__ float red[256];
  float v[4];
  float mx = -3.0e38f;
#pragma unroll
  for (int i = 0; i < 4; ++i) {
    v[i] = sim[(size_t)row * 1024 + t + i * 256];
    mx = fmaxf(mx, v[i]);
  }
  red[t] = mx; __syncthreads();
  for (int s = 128; s > 0; s >>= 1) {
    if (t < s) red[t] = fmaxf(red[t], red[t + s]);
    __syncthreads();
  }
  mx = red[0]; __syncthreads();
  float sum = 0.f;
#pragma unroll
  for (int i = 0; i < 4; ++i) { v[i] = __expf(v[i] - mx); sum += v[i]; }
  red[t] = sum; __syncthreads();
  for (int s = 128; s > 0; s >>= 1) {
    if (t < s) red[t] += red[t + s];
    __syncthreads();
  }
  const float inv = 1.0f / red[0];
#pragma unroll
  for (int i = 0; i < 4; ++i)
    sim[(size_t)row * 1024 + t + i * 256] = v[i] * inv;
}

// ---------------- K2a: sub_e[b,i,j] = exp(-L1 cdist) ------------------------
__global__ void k_cdist(const float* __restrict__ pa, float* __restrict__ sube) {
  const int b = blockIdx.z;
  const int jBase = blockIdx.x * 64, iBase = blockIdx.y * 64;
  __shared__ float Pi[64][17];   // 17-pad: conflict-free (17 coprime 64)
  __shared__ float Pj[64][17];
  const int t = threadIdx.x;
  for (int i = t; i < 64 * 16; i += 256) {
    const int r = i >> 4, d = i & 15;
    Pi[r][d] = pa[((size_t)b * 1024 + iBase + r) * 16 + d];
    Pj[r][d] = pa[((size_t)b * 1024 + jBase + r) * 16 + d];
  }
  __syncthreads();
  const int tj = t & 63;
  const int i0 = (t >> 6) * 16;
#pragma unroll 4
  for (int ii = 0; ii < 16; ++ii) {
    const int i = i0 + ii;
    float s = 0.f;
#pragma unroll
    for (int d = 0; d < 16; ++d) s += fabsf(Pi[i][d] - Pj[tj][d]);
    sube[((size_t)b * 1024 + iBase + i) * 1024 + jBase + tj] = __expf(-s);
  }
}

// ---------------- K2b: column sums S[b,j] -----------------------------------
__global__ void k_colsum(const float* __restrict__ sube, float* __restrict__ S) {
  const int b = blockIdx.y;
  const int j = blockIdx.x * 256 + threadIdx.x;
  float acc = 0.f;
  for (int i = 0; i < 1024; ++i)
    acc += sube[((size_t)b * 1024 + i) * 1024 + j];
  S[b * 1024 + j] = acc;
}

// ---------------- K2c: xn = x / S -------------------------------------------
__global__ void k_xnorm(const float* __restrict__ x, const float* __restrict__ S,
                        float* __restrict__ xn) {
  const int idx = blockIdx.x * 256 + threadIdx.x;   // < 8*1024*64
  xn[idx] = x[idx] / S[idx >> 6];
}

// ---------------- K5: W[n] = sum_d E[n,d] Wp[d], pair-interleaved -----------
// Output layout: W[n][kp][o][p] with k = 2*kp + p, i.e. element (n,k,o) at
// n*8192 + (k>>1)*128 + o*2 + (k&1).  A WMMA B-fragment {W[k][o],W[k+1][o]}
// is then one contiguous 8-byte load.
__global__ void k_weights(const float* __restrict__ E, const float* __restrict__ Wp,
                          float* __restrict__ W) {
  const int n = blockIdx.y;
  const int tid = blockIdx.x * 256 + threadIdx.x;   // < 2048
  const int kp = tid >> 5;                          // 0..63  (k pair)
  const int oc = (tid & 31) * 2;                    // 0..62  (o pair)
  float e[16];
#pragma unroll
  for (int d = 0; d < 16; ++d) e[d] = E[n * 16 + d];   // wave-uniform -> SGPR
  float4 acc = {0.f, 0.f, 0.f, 0.f};
#pragma unroll
  for (int d = 0; d < 16; ++d) {
    const v2f w0 = *(const v2f*)(Wp + (size_t)d * 8192 + (size_t)(2 * kp) * 64 + oc);
    const v2f w1 = *(const v2f*)(Wp + (size_t)d * 8192 + (size_t)(2 * kp + 1) * 64 + oc);
    acc.x += e[d] * w0.x;   // (k0, oc)
    acc.y += e[d] * w1.x;   // (k1, oc)
    acc.z += e[d] * w0.y;   // (k0, oc+1)
    acc.w += e[d] * w1.y;   // (k1, oc+1)
  }
  *(float4*)(W + (size_t)n * 8192 + (size_t)kp * 128 + (size_t)oc * 2) = acc;
}

// ---------------- K3/K4: C[b] = A[b] @ B[b], M=1024, N=64, K=1024 -----------
// WG = 256 threads = 8 waves; each wave computes a 16x64 strip via f32 WMMA.
// A staged row-major (stride 68: 68 mod 64 == 4 -> conflict-free b64 reads).
// B staged pair-interleaved Bs[k/2][n*2+(k&1)] (stride 160: 160 mod 64 == 32
// puts the two half-waves on disjoint bank halves) -> each B fragment is one
// contiguous ds_load_b64, no register shuffling before WMMA.
__global__ __launch_bounds__(256)
void k_bmm_nk(const float* __restrict__ A, long long strideA,
              const float* __restrict__ B, long long strideB,
              float* __restrict__ C, long long strideC) {
  const int b = blockIdx.y;
  const float* Ab = A + (size_t)strideA * b;
  const float* Bb = B + (size_t)strideB * b;
  float* Cb = C + (size_t)strideC * b;
  const int rowBase = blockIdx.x * 128;

  __shared__ float As[128][68];   // 34816 B
  __shared__ float Bs[32][160];   // 20480 B

  const int t = threadIdx.x;
  const int wave = t >> 5;
  const int lane = t & 31;
  const int m16 = lane & 15;
  const int kh = lane >> 4;             // 0: K={0,1}, 1: K={2,3} within k-step

  const v8f zero = {0.f, 0.f, 0.f, 0.f, 0.f, 0.f, 0.f, 0.f};
  v8f cacc[4];
#pragma unroll
  for (int nt = 0; nt < 4; ++nt) cacc[nt] = zero;

  for (int kc = 0; kc < 1024; kc += 64) {
    __syncthreads();
    // stage A chunk [128 x 64]
    for (int i = t; i < 128 * 16; i += 256) {
      const int r = i >> 4, cc = (i & 15) << 2;
      *(float4*)(&As[r][cc]) =
          *(const float4*)(Ab + (size_t)(rowBase + r) * 1024 + kc + cc);
    }
    // stage B chunk [64 x 64], pair-interleaving rows 2kp and 2kp+1
    for (int i = t; i < 32 * 16; i += 256) {
      const int kp = i >> 4, cc = (i & 15) << 2;     // 4 columns per thread
      const float4 v0 = *(const float4*)(Bb + (size_t)(kc + 2 * kp) * 64 + cc);
      const float4 v1 = *(const float4*)(Bb + (size_t)(kc + 2 * kp + 1) * 64 + cc);
      float4 lo = {v0.x, v1.x, v0.y, v1.y};
      float4 hi = {v0.z, v1.z, v0.w, v1.w};
      *(float4*)(&Bs[kp][cc * 2])     = lo;
      *(float4*)(&Bs[kp][cc * 2 + 4]) = hi;
    }
    __syncthreads();

    const int mrow = wave * 16 + m16;
#pragma unroll
    for (int ks = 0; ks < 16; ++ks) {
      const int kg = ks * 4 + 2 * kh;               // global-k within chunk
      const int kp = ks * 2 + kh;                   // pair row in Bs
      const v2f a = *(const v2f*)(&As[mrow][kg]);   // {A[m][kg], A[m][kg+1]}
#pragma unroll
      for (int nt = 0; nt < 4; ++nt) {
        const v2f bf = *(const v2f*)(&Bs[kp][(nt * 16 + m16) * 2]);
        cacc[nt] = wmma_f32(a, bf, cacc[nt]);
      }
    }
  }

  // epilogue: C rows m = r + 8*(lane/16)
#pragma unroll
  for (int nt = 0; nt < 4; ++nt) {
#pragma unroll
    for (int r = 0; r < 8; ++r) {
      const int m = rowBase + wave * 16 + r + 8 * kh;
      Cb[(size_t)m * 64 + nt * 16 + m16] = cacc[nt][r];
    }
  }
}

// ---------------- K6: final per-node mix + bias -----------------------------
// Per node n: out[0:8, n, :] = [sx|xg1] (8x128) @ Wn (128x64) + bias_n.
// One wave per node, 8 nodes per workgroup. A rows 8..15 are zero (masked).
// W is pair-interleaved, so each B fragment is one coalesced global b64 load.
__global__ __launch_bounds__(256)
void k_final(const float* __restrict__ sx, const float* __restrict__ xg1,
             const float* __restrict__ W, const float* __restrict__ E,
             const float* __restrict__ bias_pool, float* __restrict__ out) {
  __shared__ float As[8][8][132];     // [wave][row(b)][k]; 132 mod 64 == 4
  const int t = threadIdx.x;
  const int wave = t >> 5;
  const int lane = t & 31;
  const int n = blockIdx.x * 8 + wave;
  float (*Aw)[132] = As[wave];

  // stage A = concat(sx[:, n, :], xg1[:, n, :]) along K
  for (int rb = 0; rb < 8; ++rb) {
    const float* s0 = sx  + ((size_t)rb * 1024 + n) * 64;
    const float* s1 = xg1 + ((size_t)rb * 1024 + n) * 64;
    for (int c = lane; c < 128; c += 32)
      Aw[rb][c] = (c < 64) ? s0[c] : s1[c - 64];
  }
  __syncthreads();   // cross-lane LDS visibility before fragment reads

  const float* Wn = W + (size_t)n * 8192;   // pair-interleaved [64][64][2]
  const int m16 = lane & 15;
  const int kh = lane >> 4;

  const v8f zero = {0.f, 0.f, 0.f, 0.f, 0.f, 0.f, 0.f, 0.f};
  v8f cacc[4];
#pragma unroll
  for (int nt = 0; nt < 4; ++nt) cacc[nt] = zero;

#pragma unroll 4
  for (int ks = 0; ks < 32; ++ks) {
    const int kg = ks * 4 + 2 * kh;
    const int kp = ks * 2 + kh;
    v2f a = {0.f, 0.f};
    if (m16 < 8) a = *(const v2f*)(&Aw[m16][kg]);   // rows 8..15 are zero
#pragma unroll
    for (int nt = 0; nt < 4; ++nt) {
      const v2f bf = *(const v2f*)(Wn + (size_t)kp * 128 + (nt * 16 + m16) * 2);
      cacc[nt] = wmma_f32(a, bf, cacc[nt]);
    }
  }

  // bias + store (only rows 0..7 = batch index)
#pragma unroll
  for (int nt = 0; nt < 4; ++nt) {
    const int col = nt * 16 + m16;
    float bv = 0.f;
#pragma unroll
    for (int d = 0; d < 16; ++d) bv += E[n * 16 + d] * bias_pool[d * 64 + col];
#pragma unroll
    for (int r = 0; r < 8; ++r) {
      const int m = r + 8 * kh;
      if (m < 8)
        out[((size_t)m * 1024 + n) * 64 + col] = cacc[nt][r] + bv;
    }
  }
}

// ---------------------------------------------------------------------------
extern "C" void kernel_launch(void* const* d_in, const int* in_sizes, int n_in,
                              void* d_out, int out_size, void* d_ws, size_t ws_size,
                              hipStream_t stream) {
  (void)in_sizes; (void)n_in; (void)out_size; (void)ws_size;
  const float* x    = (const float*)d_in[0];   // [8,1024,64]
  const float* E    = (const float*)d_in[1];   // [1024,16]
  const float* pa   = (const float*)d_in[2];   // [8,1024,16]
  const float* Wp   = (const float*)d_in[3];   // [16,2,64,64]
  const float* bp   = (const float*)d_in[4];   // [16,64]
  float* out = (float*)d_out;                  // [8,1024,64]

  float* ws   = (float*)d_ws;
  float* sim  = ws;                    // 1,048,576
  float* sube = sim  + 1048576;        // 8,388,608
  float* S    = sube + 8388608;        //     8,192
  float* xn   = S    + 8192;           //   524,288
  float* sx   = xn   + 524288;         //   524,288
  float* xg1  = sx   + 524288;         //   524,288
  float* W    = xg1  + 524288;         // 8,388,608   (total ~77.6 MB)

  k_simraw    <<<4096, 256, 0, stream>>>(E, sim);
  k_rowsoftmax<<<1024, 256, 0, stream>>>(sim);
  k_cdist     <<<dim3(16, 16, 8), 256, 0, stream>>>(pa, sube);
  k_colsum    <<<dim3(4, 8),      256, 0, stream>>>(sube, S);
  k_xnorm     <<<2048, 256, 0, stream>>>(x, S, xn);
  k_weights   <<<dim3(8, 1024),   256, 0, stream>>>(E, Wp, W);
  // sx = sub_e @ xn   (per-batch A)
  k_bmm_nk<<<dim3(8, 8), 256, 0, stream>>>(sube, 1024LL * 1024, xn, 1024LL * 64,
                                           sx, 1024LL * 64);
  // xg1 = sim @ sx    (shared A: stride 0)
  k_bmm_nk<<<dim3(8, 8), 256, 0, stream>>>(sim, 0LL, sx, 1024LL * 64,
                                           xg1, 1024LL * 64);
  k_final<<<128, 256, 0, stream>>>(sx, xg1, W, E, bp, out);
}